// Butterfly_90658169684702
// MI455X (gfx1250) — compile-verified
//
#include <hip/hip_runtime.h>

// out = butterfly10(x) + bias  (B=32768, N=1024, fp32, increasing stride)
// WMMA formulation: the 10 butterfly stages factor exactly into two layers of
// dense 32x32 blocks:
//   layer1: stages 0-4  -> W1_c on contiguous chunks  [32c, 32c+32)
//   layer2: stages 5-9  -> W2_r on strided sets       {r, r+32, ..., r+992}
// Each layer is a batched (16x32)x(32x32) GEMM -> chains of
// V_WMMA_F32_16X16X4_F32.  HBM-bound: 268MB @ 23.3TB/s ~= 11.5us floor;
// issue pressure ~280 slots/row (vs ~750 for the VALU/pk_fma pipeline).

typedef float v2f __attribute__((ext_vector_type(2)));
typedef float v4f __attribute__((ext_vector_type(4)));
typedef float v8f __attribute__((ext_vector_type(8)));

#define YS 1028          // padded LDS row stride for Y (floats)
#define GRID 256         // workgroups in main kernel

__device__ __forceinline__ v8f wmma4(v2f a, v2f b, v8f c) {
  // D(16x16) = A(16x4) x B(4x16) + C
  return __builtin_amdgcn_wmma_f32_16x16x4_f32(false, a, false, b, (short)0, c,
                                               false, false);
}

// output-staging swizzle: quad t of 32-block q stored rotated by q (+q>>3 so
// lanes n and n+8 land on different banks); reader un-rotates with b128 reads
__device__ __forceinline__ int rotq(int r4, int q) {
  return (r4 + q + (q >> 3)) & 7;
}

// -------- kernel A: fuse 5 butterfly stages into 32x32 blocks ---------------
// wsf[blk][u][j]: blk 0..31 = W1_c (stages 0-4), blk 32..63 = W2_r (stages 5-9)
__global__ __launch_bounds__(32) void build_blocks_kernel(
    const float* __restrict__ twg, float* __restrict__ wsf)
{
  const int j   = threadIdx.x;   // input column of the block matrix
  const int blk = blockIdx.x;

  float m[32];
#pragma unroll
  for (int u = 0; u < 32; ++u) m[u] = (u == j) ? 1.0f : 0.0f;

  if (blk < 32) {
    const int c = blk;
#pragma unroll
    for (int idx = 0; idx < 5; ++idx) {
      const int s = 1 << idx;
#pragma unroll
      for (int q = 0; q < 16; ++q) {
        const int u0 = ((q >> idx) << (idx + 1)) | (q & (s - 1));
        const int u1 = u0 + s;
        // twiddle pair id: p = c*16 + q  (drop bit idx of element 32c+u0)
        v4f t = *(const v4f*)(twg + idx * 2048 + (c * 16 + q) * 4);
        float a = m[u0], b = m[u1];
        m[u0] = t.x * a + t.y * b;
        m[u1] = t.z * a + t.w * b;
      }
    }
  } else {
    const int r = blk - 32;
#pragma unroll
    for (int idx = 5; idx < 10; ++idx) {
      const int bb = idx - 5;
      const int mm = 1 << bb;
#pragma unroll
      for (int q = 0; q < 16; ++q) {
        const int c0 = ((q >> bb) << (bb + 1)) | (q & (mm - 1));
        const int c1 = c0 + mm;
        // twiddle pair id: p = q*32 + r  (drop bit idx of element 32*c0+r)
        v4f t = *(const v4f*)(twg + idx * 2048 + (q * 32 + r) * 4);
        float a = m[c0], b = m[c1];
        m[c0] = t.x * a + t.y * b;
        m[c1] = t.z * a + t.w * b;
      }
    }
  }

  float* w = wsf + blk * 1024 + j;
#pragma unroll
  for (int u = 0; u < 32; ++u) w[u * 32] = m[u];   // W[blk][u][j], coalesced
}

// -------- main kernel: two WMMA layers per 16-row batch tile ----------------
__global__ __launch_bounds__(256) void butterfly_wmma_kernel(
    const float* __restrict__ x, const float* __restrict__ wsf,
    const float* __restrict__ bias, float* __restrict__ out, int tpw)
{
  __shared__ __align__(16) float sY[16 * YS];     // layer1 -> layer2 tile
  __shared__ __align__(16) float sO[16 * 1024];   // swizzled output tile

  const int lane = threadIdx.x & 31;
  const int wave = threadIdx.x >> 5;
  const int nlo  = lane & 15;       // N (or M) index within WMMA tile
  const int hi   = lane >> 4;       // lane-half: K offset selector

  // ---- per-wave invariant operands: 4 c-blocks + 4 r-blocks of W, in VGPRs.
  // B operand (4x16 f32): VGPR0 = {K=k0 (lanes0-15), K=k0+2 (lanes16-31)},
  // VGPR1 = K+1 -> one b64 load per K-chunk: B[kk].xy = W[n, k0 + hi*2 + 0/1]
  v2f B1[4][2][8], B2[4][2][8];
  float bsr[4][2];
#pragma unroll
  for (int bi = 0; bi < 4; ++bi) {
    const int c = wave * 4 + bi;    // also the r-block index for layer 2
#pragma unroll
    for (int nt = 0; nt < 2; ++nt) {
      const int n = nt * 16 + nlo;
#pragma unroll
      for (int kk = 0; kk < 8; ++kk) {
        B1[bi][nt][kk] = *(const v2f*)(wsf + c * 1024 + n * 32 + kk * 4 + hi * 2);
        B2[bi][nt][kk] = *(const v2f*)(wsf + 32768 + c * 1024 + n * 32 + kk * 4 + hi * 2);
      }
      bsr[bi][nt] = bias[32 * n + c];   // bias for out col 32*n + r, r = c
    }
  }

#pragma unroll 1
  for (int tt = 0; tt < tpw; ++tt) {
    const int tile = blockIdx.x * tpw + tt;
    const size_t row0 = (size_t)tile * 16;

    // ---------------- layer 1: Y = X * W1^T (blockwise) ----------------
#pragma unroll
    for (int bi = 0; bi < 4; ++bi) {
      const int c = wave * 4 + bi;
      // A operand (16x4 f32): lane = M (batch row), VGPR+half = K
      const float* ab = x + (row0 + nlo) * 1024 + c * 32 + hi * 2;
      v2f A[8];
#pragma unroll
      for (int kk = 0; kk < 8; ++kk) A[kk] = *(const v2f*)(ab + kk * 4);
#pragma unroll
      for (int nt = 0; nt < 2; ++nt) {
        v8f acc = {0.f, 0.f, 0.f, 0.f, 0.f, 0.f, 0.f, 0.f};
#pragma unroll
        for (int kk = 0; kk < 8; ++kk) acc = wmma4(A[kk], B1[bi][nt][kk], acc);
        // D: VGPR v -> M = v + 8*hi, N = nlo
#pragma unroll
        for (int v = 0; v < 8; ++v)
          sY[(v + 8 * hi) * YS + c * 32 + nt * 16 + nlo] = acc[v];
      }
    }
    __syncthreads();

    // ---------------- layer 2: out = Y_r * W2_r^T (blockwise) ----------
#pragma unroll
    for (int bi = 0; bi < 4; ++bi) {
      const int r = wave * 4 + bi;
      v2f A2[8];
#pragma unroll
      for (int kk = 0; kk < 8; ++kk) {
        const int K = kk * 4 + hi * 2;
        A2[kk].x = sY[nlo * YS + 32 * K + r];
        A2[kk].y = sY[nlo * YS + 32 * (K + 1) + r];
      }
#pragma unroll
      for (int nt = 0; nt < 2; ++nt) {
        v8f acc = {0.f, 0.f, 0.f, 0.f, 0.f, 0.f, 0.f, 0.f};
#pragma unroll
        for (int kk = 0; kk < 8; ++kk) acc = wmma4(A2[kk], B2[bi][nt][kk], acc);
        const int q = nt * 16 + nlo;               // out col = 32q + r
#pragma unroll
        for (int v = 0; v < 8; ++v) {
          float val = acc[v] + bsr[bi][nt];
          sO[(v + 8 * hi) * 1024 + 32 * q + 4 * rotq(r >> 2, q) + (r & 3)] = val;
        }
      }
    }
    __syncthreads();

    // -------- coalesced copy-out: wave w streams rows 2w, 2w+1 ----------
    // (overlaps next tile's layer 1; next tile's first barrier fences sO)
#pragma unroll
    for (int mi = 0; mi < 2; ++mi) {
      const int m = wave * 2 + mi;
      float* orow = out + (row0 + m) * 1024;
#pragma unroll
      for (int jj = 0; jj < 8; ++jj) {
        const int ch = jj * 32 + lane;
        const int q = ch >> 3, t = ch & 7;
        v4f v = *(const v4f*)(sO + m * 1024 + 32 * q + 4 * rotq(t, q));
        __builtin_nontemporal_store(v, (v4f*)(orow + 32 * q + 4 * t));
      }
    }
  }
}

extern "C" void kernel_launch(void* const* d_in, const int* in_sizes, int n_in,
                              void* d_out, int out_size, void* d_ws, size_t ws_size,
                              hipStream_t stream) {
  const float* x    = (const float*)d_in[0];  // (32768, 1024)
  const float* twg  = (const float*)d_in[1];  // (1, 10, 512, 2, 2)
  const float* bias = (const float*)d_in[2];  // (1024,)
  float* out = (float*)d_out;
  float* wsf = (float*)d_ws;                  // needs 64*32*32*4 = 256KB

  build_blocks_kernel<<<dim3(64), dim3(32), 0, stream>>>(twg, wsf);

  const int rows  = in_sizes[0] / 1024;       // 32768
  const int tiles = rows / 16;                // 2048
  const int tpw   = tiles / GRID;             // 8 tiles per workgroup
  butterfly_wmma_kernel<<<dim3(GRID), dim3(256), 0, stream>>>(x, wsf, bias, out, tpw);
}